// Attention_38448547234515
// MI455X (gfx1250) — compile-verified
//
#include <hip/hip_runtime.h>
#include <hip/hip_bf16.h>

typedef float v2f __attribute__((ext_vector_type(2)));
typedef float v8f __attribute__((ext_vector_type(8)));

#define B_ 16
#define L_ 512
#define I_ 300
#define H_ 32
#define O_ 15

// ---------------------------------------------------------------------------
// C[M,N] = A[M,K] @ W[N,K]^T + bias[N]   (torch Linear layout), f32 WMMA.
// One 16x16 C tile per wave via V_WMMA_F32_16X16X4_F32, K consumed 4/step.
// A frag (16x4): lanes 0-15 rows M=0..15; VGPR0 holds K = k0+2*(lane>>4),
// VGPR1 holds K+1.  B frag (4x16) mirrors it with B[k][n] = W[n][k], so the
// per-lane load pattern for W (row-major [N,K]) is identical to A.
// D (16x16 f32): lane%16 = N, VGPR r -> M = r + 8*(lane>>4).
// Requires M%64==0, N%16==0, K%4==0 (true for all call sites here).
// ---------------------------------------------------------------------------
__global__ void gemm_wmma_f32(const float* __restrict__ A,
                              const float* __restrict__ W,
                              const float* __restrict__ bias,
                              float* __restrict__ C,
                              int M, int N, int K) {
  const int lane = threadIdx.x & 31;
  const int wave = threadIdx.x >> 5;
  const int m0 = (blockIdx.x * 4 + wave) * 16;
  const int n0 = blockIdx.y * 16;
  if (m0 >= M) return;

  const int row = lane & 15;
  const int kg  = lane >> 4;                 // 0 or 1: which K-pair this lane holds
  const float* __restrict__ arow = A + (size_t)(m0 + row) * K + 2 * kg;
  const float* __restrict__ wrow = W + (size_t)(n0 + row) * K + 2 * kg;

  v8f acc = {0.f, 0.f, 0.f, 0.f, 0.f, 0.f, 0.f, 0.f};
  for (int k0 = 0; k0 < K; k0 += 4) {
    v2f a, b;
    a.x = arow[k0];
    a.y = arow[k0 + 1];
    b.x = wrow[k0];
    b.y = wrow[k0 + 1];
    // 8 args: (neg_a, A, neg_b, B, c_mod, C, reuse_a, reuse_b)
    acc = __builtin_amdgcn_wmma_f32_16x16x4_f32(
        false, a, false, b, (short)0, acc, false, false);
  }

  const float bn = bias ? bias[n0 + row] : 0.f;
#pragma unroll
  for (int r = 0; r < 8; ++r) {
    const int m = m0 + r + 8 * kg;
    C[(size_t)m * N + (n0 + row)] = acc[r] + bn;
  }
}

// ---------------------------------------------------------------------------
// Sequential GRU scan. One block (1 wave, 32 lanes) per batch element; lane j
// owns hidden unit j.  w_hh rows for (r,z,n) gates live in VGPRs; h is
// broadcast through a 32-float LDS buffer each step.  Output is masked by
// x_lens (h itself keeps evolving past the mask, matching jax.lax.scan).
// ---------------------------------------------------------------------------
__global__ void gru_kernel(const float* __restrict__ gx,    // [B*L, 3H] = x@w_ih^T + b_ih
                           const float* __restrict__ w_hh,  // [3H, H]
                           const float* __restrict__ b_hh,  // [3H]
                           const float* __restrict__ h0,    // [1,B,H]
                           const int*   __restrict__ x_lens,// [B]
                           float* __restrict__ out)         // [B,L,H]
{
  const int b = blockIdx.x;
  const int j = threadIdx.x;                 // 0..31
  __shared__ float hbuf[H_];

  float wr[H_], wz[H_], wn_[H_];
#pragma unroll
  for (int k = 0; k < H_; ++k) {
    wr[k]  = w_hh[(0 * H_ + j) * H_ + k];
    wz[k]  = w_hh[(1 * H_ + j) * H_ + k];
    wn_[k] = w_hh[(2 * H_ + j) * H_ + k];
  }
  const float br = b_hh[j], bz = b_hh[H_ + j], bn = b_hh[2 * H_ + j];

  float h = h0[b * H_ + j];
  hbuf[j] = h;
  __syncthreads();
  const int len = x_lens[b];

  for (int t = 0; t < L_; ++t) {
    float rh = br, zh = bz, nh = bn;
#pragma unroll
    for (int k = 0; k < H_; ++k) {
      const float hk = hbuf[k];
      rh += wr[k] * hk;
      zh += wz[k] * hk;
      nh += wn_[k] * hk;
    }
    const float* __restrict__ g = gx + (size_t)(b * L_ + t) * (3 * H_);
    const float rx = g[j], zx = g[H_ + j], nx = g[2 * H_ + j];
    const float r = 1.f / (1.f + expf(-(rx + rh)));
    const float z = 1.f / (1.f + expf(-(zx + zh)));
    const float n = tanhf(nx + r * nh);
    const float hnew = (1.f - z) * n + z * h;
    out[(size_t)(b * L_ + t) * H_ + j] = (t < len) ? hnew : 0.f;
    h = hnew;
    __syncthreads();
    hbuf[j] = h;
    __syncthreads();
  }
}

// ---------------------------------------------------------------------------
// Additive attention + log_softmax + context + output head.
// One wave per output row (b,i); lane = hidden index h.
//   e[j]   = sum_h v[h] * tanh(qh[b,i,h] + kh[b,j,h])   (xor-reduce per j)
//   att    = e - max - log(sum exp)                     (log_softmax weights)
//   ctx[h] = sum_j att[j] * out[b,j,h]                  (coalesced loads)
//   y[o]   = fc_w[o,:] . ctx + fc_b[o]                  (lanes 0..14)
// Each wave owns a private LDS slice; no cross-wave synchronization.
// ---------------------------------------------------------------------------
#define ATT_WAVES 8
__global__ void att_kernel(const float* __restrict__ qh,    // [B*L, H]
                           const float* __restrict__ kh,    // [B*L, H]
                           const float* __restrict__ outg,  // [B*L, H] (masked GRU out)
                           const float* __restrict__ v,     // [H]
                           const float* __restrict__ fc_w,  // [O, H]
                           const float* __restrict__ fc_b,  // [O]
                           float* __restrict__ y)           // [B*L, O]
{
  __shared__ float e_s[ATT_WAVES][L_];
  __shared__ float ctx_s[ATT_WAVES][H_];
  const int wave = threadIdx.x >> 5;
  const int lane = threadIdx.x & 31;
  const int ri = blockIdx.x * ATT_WAVES + wave;  // row in [0, B*L)
  const int b = ri / L_;

  const float q  = qh[(size_t)ri * H_ + lane];
  const float vv = v[lane];
  const float* __restrict__ khb = kh + (size_t)b * L_ * H_;
  float* __restrict__ ew = e_s[wave];

  for (int jj = 0; jj < L_; ++jj) {
    float t = tanhf(q + khb[(size_t)jj * H_ + lane]) * vv;
#pragma unroll
    for (int off = 16; off > 0; off >>= 1) t += __shfl_xor(t, off, 32);
    if (lane == 0) ew[jj] = t;
  }

  // log-softmax over ew[0..L-1]
  float mloc = -3.402823466e38f;
  for (int jj = lane; jj < L_; jj += 32) mloc = fmaxf(mloc, ew[jj]);
#pragma unroll
  for (int off = 16; off > 0; off >>= 1)
    mloc = fmaxf(mloc, __shfl_xor(mloc, off, 32));
  float sloc = 0.f;
  for (int jj = lane; jj < L_; jj += 32) sloc += expf(ew[jj] - mloc);
#pragma unroll
  for (int off = 16; off > 0; off >>= 1) sloc += __shfl_xor(sloc, off, 32);
  const float logz = mloc + logf(sloc);
  for (int jj = lane; jj < L_; jj += 32) ew[jj] -= logz;

  // ctx[h] = sum_j att[j] * out[b,j,h]
  const float* __restrict__ ob = outg + (size_t)b * L_ * H_;
  float acc = 0.f;
  for (int jj = 0; jj < L_; ++jj) acc += ew[jj] * ob[(size_t)jj * H_ + lane];
  ctx_s[wave][lane] = acc;

  // output head, O=15
  if (lane < O_) {
    float yv = fc_b[lane];
#pragma unroll
    for (int k = 0; k < H_; ++k) yv += fc_w[lane * H_ + k] * ctx_s[wave][k];
    y[(size_t)ri * O_ + lane] = yv;
  }
}

// ---------------------------------------------------------------------------
extern "C" void kernel_launch(void* const* d_in, const int* in_sizes, int n_in,
                              void* d_out, int out_size, void* d_ws, size_t ws_size,
                              hipStream_t stream) {
  const float* x      = (const float*)d_in[0];   // [B,L,I]
  const int*   x_lens = (const int*)  d_in[1];   // [B]
  const float* h0     = (const float*)d_in[2];   // [1,B,H]
  const float* w_ih   = (const float*)d_in[3];   // [3H,I]
  const float* w_hh   = (const float*)d_in[4];   // [3H,H]
  const float* b_ih   = (const float*)d_in[5];   // [3H]
  const float* b_hh   = (const float*)d_in[6];   // [3H]
  const float* fc_w   = (const float*)d_in[7];   // [O,H]
  const float* fc_b   = (const float*)d_in[8];   // [O]
  const float* fch_w  = (const float*)d_in[9];   // [H,H]
  const float* fco_w  = (const float*)d_in[10];  // [H,H]
  const float* vvec   = (const float*)d_in[11];  // [H]

  const int M = B_ * L_;                         // 8192 rows
  float* ws   = (float*)d_ws;
  float* gx   = ws;                              // [M, 3H]  = 786432 f
  float* gout = gx   + (size_t)M * 3 * H_;       // [M, H]   = 262144 f
  float* qh   = gout + (size_t)M * H_;           // [M, H]
  float* kh   = qh   + (size_t)M * H_;           // [M, H]   (total 6 MB)

  // 1) gx = x @ w_ih^T + b_ih   (WMMA f32: M=8192, N=96, K=300)
  gemm_wmma_f32<<<dim3(M / 64, (3 * H_) / 16), dim3(128), 0, stream>>>(
      x, w_ih, b_ih, gx, M, 3 * H_, I_);

  // 2) sequential GRU scan (+ length masking)
  gru_kernel<<<dim3(B_), dim3(32), 0, stream>>>(gx, w_hh, b_hh, h0, x_lens, gout);

  // 3) qh = out @ fch_w^T ; kh = out @ fco_w^T  (WMMA f32: N=32, K=32)
  gemm_wmma_f32<<<dim3(M / 64, H_ / 16), dim3(128), 0, stream>>>(
      gout, fch_w, nullptr, qh, M, H_, H_);
  gemm_wmma_f32<<<dim3(M / 64, H_ / 16), dim3(128), 0, stream>>>(
      gout, fco_w, nullptr, kh, M, H_, H_);

  // 4) attention + log_softmax + context + head
  att_kernel<<<dim3(M / ATT_WAVES), dim3(ATT_WAVES * 32), 0, stream>>>(
      qh, kh, gout, vvec, fc_w, fc_b, (float*)d_out);
}